// SFNN_84516366450978
// MI455X (gfx1250) — compile-verified
//
#include <hip/hip_runtime.h>
#include <hip/hip_bf16.h>

typedef float v2f __attribute__((ext_vector_type(2)));
typedef float v8f __attribute__((ext_vector_type(8)));

#define NI 64
#define NH 1408
#define NO 64
#define NN 1536
#define ND 8
#define NG 24

#define NP_OFF 1
#define HID_OFF (NP_OFF + NN * ND)

// workspace layout (float offsets)
#define WS_COLSUM1  0                               // NH*8   = 11264
#define WS_COLSUM2  (WS_COLSUM1 + NH * ND)          // +11776 (atomics; zeroed)
#define WS_POST_IN  (WS_COLSUM2 + (NH + NO) * ND)   // 23040
#define WS_POST_HID (WS_POST_IN + NI * ND)
#define WS_POST_OUT (WS_POST_HID + NH * ND)
#define WS_ROWGI1   (WS_POST_OUT + NO * ND)
#define WS_COLGI1   (WS_ROWGI1 + NI * NG)
#define WS_ROWGI2   (WS_COLGI1 + NH * NG)
#define WS_COLGI2   (WS_ROWGI2 + NH * NG)
#define WS_ROWGI3   (WS_COLGI2 + (NH + NO) * NG)
#define WS_COLGI3   (WS_ROWGI3 + NI * NG)
// total = 175104 floats (~700 KB)

// fast reciprocal (v_rcp_f32) instead of IEEE divide (v_div_scale/v_div_fmas chain)
__device__ __forceinline__ float frcp(float x) {
    return __builtin_amdgcn_rcpf(x);
}
__device__ __forceinline__ float fsigmoid(float x) {
    return frcp(1.0f + __expf(-x));
}
__device__ __forceinline__ float ftanh(float x) {
    float ax = fabsf(x);
    float e  = __expf(-2.0f * ax);
    float t  = (1.0f - e) * frcp(1.0f + e);
    return copysignf(t, x);
}

// ---------------- copy unchanged hidden blocks ----------------
// unchanged cells: [all rows, cols 0..63], [rows 0..63, cols 1472..1535],
//                  [rows 1472..1535, cols 1472..1535]
__global__ void k_copy(const float* __restrict__ hin, float* __restrict__ hout) {
    int f = blockIdx.x * 256 + threadIdx.x;
    const int total = (NN * 64 + 64 * 64 + 64 * 64) * ND;  // 851968
    if (f >= total) return;
    int cellf = f >> 3, j = f & 7;
    int r, c;
    if (cellf < NN * 64) {
        r = cellf >> 6; c = cellf & 63;
    } else {
        int q = cellf - NN * 64;
        if (q < 64 * 64) { r = q >> 6; c = 1472 + (q & 63); }
        else { q -= 64 * 64; r = 1472 + (q >> 6); c = 1472 + (q & 63); }
    }
    size_t idx = ((size_t)r * NN + c) * ND + j;
    hout[idx] = hin[idx];
}

// ---------------- post_in = tanh(obs[i]*rowsum(nW_in[j]) + nb_in[j]) ----------------
__global__ void k_post_in(const float* __restrict__ obs, const float* __restrict__ W,
                          const float* __restrict__ b, float* __restrict__ post,
                          float* __restrict__ out_np) {
    int t = blockIdx.x * blockDim.x + threadIdx.x;
    if (t >= NI * ND) return;
    int i = t >> 3, j = t & 7;
    float s = 0.f;
#pragma unroll
    for (int k = 0; k < 8; ++k) s += W[j * 8 + k];
    float v = ftanh(obs[i] * s + b[j]);
    post[t] = v;
    out_np[t] = v;
}

// ---------------- post = tanh(sum @ W.T + b) ----------------
__global__ void k_post_mat(const float* __restrict__ sum, const float* __restrict__ W,
                           const float* __restrict__ b, float* __restrict__ post,
                           float* __restrict__ out_np, int Rn) {
    int t = blockIdx.x * blockDim.x + threadIdx.x;
    if (t >= Rn * ND) return;
    int i = t >> 3, j = t & 7;
    float s = b[j];
#pragma unroll
    for (int k = 0; k < 8; ++k) s += sum[i * 8 + k] * W[j * 8 + k];
    float v = ftanh(s);
    post[t] = v;
    out_np[t] = v;
}

// ---------------- action = argmax(post_out[:,0]) ----------------
__global__ void k_action(const float* __restrict__ post_out, float* __restrict__ out) {
    if (threadIdx.x == 0 && blockIdx.x == 0) {
        float best = post_out[0];
        int bi = 0;
        for (int i = 1; i < NO; ++i) {
            float v = post_out[i * ND];
            if (v > best) { best = v; bi = i; }
        }
        out[0] = (float)bi;
    }
}

// ---------------- rowGI[i][g] = Wih[g,0:8] . pre[i] ----------------
__global__ void k_gi_rows(const float* __restrict__ W, const float* __restrict__ pre,
                          float* __restrict__ outg, int R) {
    int t = blockIdx.x * blockDim.x + threadIdx.x;
    if (t >= R * NG) return;
    int i = t / NG, g = t % NG;
    float s = 0.f;
#pragma unroll
    for (int k = 0; k < 8; ++k) s += W[g * 17 + k] * pre[(size_t)i * ND + k];
    outg[t] = s;
}

// ---------------- colGI[i][g] = Wih[g,8:16] . post[i] + Wih[g,16]*reward + bih[g] --
__global__ void k_gi_cols(const float* __restrict__ W, const float* __restrict__ bih,
                          const float* __restrict__ rew,
                          const float* __restrict__ pA, int lenA,
                          const float* __restrict__ pB,
                          float* __restrict__ outg, int C) {
    int t = blockIdx.x * blockDim.x + threadIdx.x;
    if (t >= C * NG) return;
    int i = t / NG, g = t % NG;
    const float* p = (i < lenA) ? (pA + (size_t)i * ND) : (pB + (size_t)(i - lenA) * ND);
    float s = bih[g] + W[g * 17 + 16] * rew[0];
#pragma unroll
    for (int k = 0; k < 8; ++k) s += W[g * 17 + 8 + k] * p[k];
    outg[t] = s;
}

// ---------------- main plastic-GRU stage ----------------
// Wave tile: 1 row x 16 cols. gh via V_WMMA_F32_16X16X4_F32:
//   A (16x4 f32): lane l<16 -> M=l, K=kk..kk+1 with kk=0; l>=16 -> K=2,3  (2 VGPRs)
//   B (4x16 f32): lane l -> N=l%16, K=2*(l/16)(+1)                       (2 VGPRs)
//   D (16x16 f32): lane l, vgpr v -> M=v+8*(l/16), N=l%16                (8 VGPRs)
__global__ __launch_bounds__(256)
void stage_kernel(const float* __restrict__ hid_in,
                  float* __restrict__ hid_out,
                  const unsigned char* __restrict__ adj,
                  const float* __restrict__ rowGI,
                  const float* __restrict__ colGI,
                  const float* __restrict__ pre,
                  const float* __restrict__ whh,   // 24x8
                  const float* __restrict__ bhh,   // 24
                  const float* __restrict__ lr_p,
                  float* __restrict__ colsum,      // C x 8 (atomic) or nullptr
                  int R, int r0g, int c0g,
                  int rows_per_wave, int colsum_min_col) {
    __shared__ float s_colGI[16 * NG];
    __shared__ float s_gh[8][16 * NG];
    __shared__ float s_h[8][16 * ND];

    const int lane = threadIdx.x & 31;
    const int wave = threadIdx.x >> 5;
    const int c_base = blockIdx.x * 16;                    // local col base
    const int chunk_base = blockIdx.y * (8 * rows_per_wave);

    for (int t = threadIdx.x; t < 16 * NG; t += 256) {
        int cc = t / NG, g = t % NG;
        s_colGI[t] = colGI[(size_t)(c_base + cc) * NG + g];
    }
    __syncthreads();

    const float lrv = lr_p[0];

    // B tiles (Whh^T), loop invariant
    const int kk  = (lane >> 4) * 2;  // 0 or 2
    const int m16 = lane & 15;
    const int g1  = lane & 15;
    const int g2  = 16 + (lane & 15);
    v2f B00 = *(const v2f*)(whh + g1 * ND + kk);
    v2f B01 = *(const v2f*)(whh + g1 * ND + 4 + kk);
    v2f B10 = {0.f, 0.f}, B11 = {0.f, 0.f};
    if (g2 < NG) {
        B10 = *(const v2f*)(whh + g2 * ND + kk);
        B11 = *(const v2f*)(whh + g2 * ND + 4 + kk);
    }

    // per-lane (cell,d) assignments for elementwise phase: 4 pairs, fixed across rows
    int cl[4], dl[4];
    float cgr[4], cgz[4], cgn[4], bhr[4], bhz[4], bhn[4], acc[4];
#pragma unroll
    for (int t = 0; t < 4; ++t) {
        int idx = t * 32 + lane;
        int cell = idx >> 3, d = idx & 7;
        cl[t] = cell; dl[t] = d;
        cgr[t] = s_colGI[cell * NG + d];
        cgz[t] = s_colGI[cell * NG + 8 + d];
        cgn[t] = s_colGI[cell * NG + 16 + d];
        bhr[t] = bhh[d]; bhz[t] = bhh[8 + d]; bhn[t] = bhh[16 + d];
        acc[t] = 0.f;
    }

    const int cellD = (lane >> 4) * 8;
    const int gateD = lane & 15;

    for (int i = 0; i < rows_per_wave; ++i) {
        const int r = chunk_base + wave + 8 * i;   // uniform within wave
        if (r >= R) break;
        const size_t rowcell = (size_t)(r0g + r) * NN + (c0g + c_base);
        const float* hrow = hid_in + rowcell * ND;

        if (i + 1 < rows_per_wave) {
            const float* nx = hid_in + ((size_t)(r0g + r + 8) * NN + (c0g + c_base)) * ND;
            __builtin_prefetch((const char*)nx + lane * 16, 0, 3);
        }

        // A tile: h[r, c_base..c_base+15, 0..7]  (coalesced 512B)
        v2f A0 = *(const v2f*)(hrow + m16 * ND + kk);
        v2f A1 = *(const v2f*)(hrow + m16 * ND + 4 + kk);
        s_h[wave][m16 * ND + kk]     = A0.x;
        s_h[wave][m16 * ND + kk + 1] = A0.y;
        s_h[wave][m16 * ND + 4 + kk]     = A1.x;
        s_h[wave][m16 * ND + 4 + kk + 1] = A1.y;

        // gh = h * Whh^T  (16 cells x 24 gates), K=8 split into two K=4 WMMAs
        v8f d0 = {}; v8f d1 = {};
        d0 = __builtin_amdgcn_wmma_f32_16x16x4_f32(false, A0, false, B00, (short)0, d0, false, false);
        d0 = __builtin_amdgcn_wmma_f32_16x16x4_f32(false, A1, false, B01, (short)0, d0, false, false);
        d1 = __builtin_amdgcn_wmma_f32_16x16x4_f32(false, A0, false, B10, (short)0, d1, false, false);
        d1 = __builtin_amdgcn_wmma_f32_16x16x4_f32(false, A1, false, B11, (short)0, d1, false, false);

#pragma unroll
        for (int v = 0; v < 8; ++v)
            s_gh[wave][(cellD + v) * NG + gateD] = d0[v];
        if (gateD < 8) {
#pragma unroll
            for (int v = 0; v < 8; ++v)
                s_gh[wave][(cellD + v) * NG + 16 + gateD] = d1[v];
        }

        // GRU elementwise: 16 cells x 8 d = 128 items, 4 per lane
        const float* rg = rowGI + (size_t)r * NG;
#pragma unroll
        for (int t = 0; t < 4; ++t) {
            const int cell = cl[t], d = dl[t];
            const float hv = s_h[wave][cell * ND + d];
            const float hr = s_gh[wave][cell * NG + d] + bhr[t];
            const float hz = s_gh[wave][cell * NG + 8 + d] + bhz[t];
            const float hn = s_gh[wave][cell * NG + 16 + d] + bhn[t];
            const float rr = fsigmoid(rg[d] + cgr[t] + hr);
            const float zz = fsigmoid(rg[8 + d] + cgz[t] + hz);
            const float nn = ftanh(rg[16 + d] + cgn[t] + rr * hn);
            const float g  = (1.f - zz) * nn + zz * hv;
            const unsigned char mk = adj[rowcell + cell];
            const float hnew = mk ? (hv + g * lrv) : hv;
            hid_out[(rowcell + cell) * ND + d] = hnew;
            if (mk) acc[t] += hnew * pre[(size_t)r * ND + d];
        }
    }

    if (colsum != nullptr) {
#pragma unroll
        for (int t = 0; t < 4; ++t) {
            const int c = c_base + cl[t];
            if (c >= colsum_min_col)
                atomicAdd(&colsum[(size_t)c * ND + dl[t]], acc[t]);
        }
    }
}

extern "C" void kernel_launch(void* const* d_in, const int* in_sizes, int n_in,
                              void* d_out, int out_size, void* d_ws, size_t ws_size,
                              hipStream_t stream) {
    (void)in_sizes; (void)n_in; (void)out_size; (void)ws_size;

    const float* obs   = (const float*)d_in[0];
    const float* rew   = (const float*)d_in[1];
    const float* lr    = (const float*)d_in[2];
    const float* post0 = (const float*)d_in[3];
    const float* hid0  = (const float*)d_in[4];
    const unsigned char* adj = (const unsigned char*)d_in[5];
    const float* nW_in  = (const float*)d_in[6];
    const float* nb_in  = (const float*)d_in[7];
    const float* nW_hid = (const float*)d_in[8];
    const float* nb_hid = (const float*)d_in[9];
    const float* nW_out = (const float*)d_in[10];
    const float* nb_out = (const float*)d_in[11];
    const float* sWih_in  = (const float*)d_in[12];
    const float* sWhh_in  = (const float*)d_in[13];
    const float* sbih_in  = (const float*)d_in[14];
    const float* sbhh_in  = (const float*)d_in[15];
    const float* sWih_hid = (const float*)d_in[16];
    const float* sWhh_hid = (const float*)d_in[17];
    const float* sbih_hid = (const float*)d_in[18];
    const float* sbhh_hid = (const float*)d_in[19];
    const float* sWih_out = (const float*)d_in[20];
    const float* sWhh_out = (const float*)d_in[21];
    const float* sbih_out = (const float*)d_in[22];
    const float* sbhh_out = (const float*)d_in[23];

    float* out = (float*)d_out;
    float* ws  = (float*)d_ws;

    float* colsum1  = ws + WS_COLSUM1;
    float* colsum2  = ws + WS_COLSUM2;
    float* post_in  = ws + WS_POST_IN;
    float* post_hid = ws + WS_POST_HID;
    float* post_out = ws + WS_POST_OUT;
    float* rowGI1 = ws + WS_ROWGI1;
    float* colGI1 = ws + WS_COLGI1;
    float* rowGI2 = ws + WS_ROWGI2;
    float* colGI2 = ws + WS_COLGI2;
    float* rowGI3 = ws + WS_ROWGI3;
    float* colGI3 = ws + WS_COLGI3;

    // zero atomic column-sum accumulators
    hipMemsetAsync(d_ws, 0, (size_t)WS_POST_IN * sizeof(float), stream);

    // copy unchanged hidden blocks into output
    k_copy<<<3328, 256, 0, stream>>>(hid0, out + HID_OFF);

    // post_in
    k_post_in<<<2, 256, 0, stream>>>(obs, nW_in, nb_in, post_in, out + NP_OFF);

    // stage 1: rows 0..63, cols 64..1471
    k_gi_rows<<<(NI * NG + 255) / 256, 256, 0, stream>>>(sWih_in, post_in, rowGI1, NI);
    k_gi_cols<<<(NH * NG + 255) / 256, 256, 0, stream>>>(sWih_in, sbih_in, rew,
                                                         post0 + NI * ND, NH, nullptr,
                                                         colGI1, NH);
    stage_kernel<<<dim3(NH / 16, 1), 256, 0, stream>>>(hid0, out + HID_OFF, adj,
                                                       rowGI1, colGI1, post_in,
                                                       sWhh_in, sbhh_in, lr, colsum1,
                                                       NI, 0, NI, 8, 0);

    // post_hid
    k_post_mat<<<(NH * ND + 255) / 256, 256, 0, stream>>>(colsum1, nW_hid, nb_hid,
                                                          post_hid, out + NP_OFF + NI * ND, NH);

    // stage 2: rows 64..1471, cols 64..1535
    k_gi_rows<<<(NH * NG + 255) / 256, 256, 0, stream>>>(sWih_hid, post_hid, rowGI2, NH);
    k_gi_cols<<<((NH + NO) * NG + 255) / 256, 256, 0, stream>>>(sWih_hid, sbih_hid, rew,
                                                                post_hid, NH,
                                                                post0 + (NI + NH) * ND,
                                                                colGI2, NH + NO);
    stage_kernel<<<dim3((NH + NO) / 16, 11), 256, 0, stream>>>(hid0, out + HID_OFF, adj,
                                                               rowGI2, colGI2, post_hid,
                                                               sWhh_hid, sbhh_hid, lr, colsum2,
                                                               NH, NI, NI, 16, NH);

    // post_out + action
    k_post_mat<<<2, 256, 0, stream>>>(colsum2 + NH * ND, nW_out, nb_out,
                                      post_out, out + NP_OFF + (NI + NH) * ND, NO);
    k_action<<<1, 32, 0, stream>>>(post_out, out);

    // stage 3: rows 1472..1535, cols 64..1471 (no colsum needed)
    k_gi_rows<<<(NI * NG + 255) / 256, 256, 0, stream>>>(sWih_out, post_out, rowGI3, NI);
    k_gi_cols<<<(NH * NG + 255) / 256, 256, 0, stream>>>(sWih_out, sbih_out, rew,
                                                         post_hid, NH, nullptr,
                                                         colGI3, NH);
    stage_kernel<<<dim3(NH / 16, 1), 256, 0, stream>>>(hid0, out + HID_OFF, adj,
                                                       rowGI3, colGI3, post_out,
                                                       sWhh_out, sbhh_out, lr, nullptr,
                                                       NI, NI + NH, NI, 8, NH);
}